// MultiplePairsContrastiveLoss_23974507446579
// MI455X (gfx1250) — compile-verified
//
#include <hip/hip_runtime.h>
#include <hip/hip_bf16.h>
#include <math.h>

typedef __attribute__((ext_vector_type(2))) float v2f;
typedef __attribute__((ext_vector_type(8))) float v8f;

#define N_ROWS 4096
#define D_FEAT 512
#define D_CONT 14
#define K_CAT  336
#define N_MT   (N_ROWS / 32)          // 128 macro-tiles (32x32) per dim
#define SIM_THR 0.05f
#define INV_TEMP 10.0f
#define LAM 0.5f

static __device__ __forceinline__ v8f wmma_f32_4(v2f a, v2f b, v8f c) {
    // D = A(16x4,f32) * B(4x16,f32) + C(16x16,f32)
    return __builtin_amdgcn_wmma_f32_16x16x4_f32(
        /*neg_a=*/false, a, /*neg_b=*/false, b,
        /*c_mod=*/(short)0, c, /*reuse_a=*/false, /*reuse_b=*/false);
}

// ---------------------------------------------------------------------------
// Kernel 1: per-row prep. One wave (32 lanes) per row.
// ---------------------------------------------------------------------------
__global__ void prep_kernel(const float* __restrict__ feat,
                            const float* __restrict__ cont,
                            float* __restrict__ fn,
                            float* __restrict__ csum,
                            float* __restrict__ pos,
                            float* __restrict__ den) {
    const int lane = threadIdx.x & 31;
    const int wave = threadIdx.x >> 5;
    const int row  = blockIdx.x * 8 + wave;
    if (row >= N_ROWS) return;

    float s = 0.f;
    for (int k = lane; k < D_FEAT; k += 32) {
        float v = feat[(size_t)row * D_FEAT + k];
        s += v * v;
    }
    #pragma unroll
    for (int o = 16; o >= 1; o >>= 1) s += __shfl_xor(s, o, 32);
    const float invn = rsqrtf(s);

    for (int k = lane; k < D_FEAT; k += 32)
        fn[(size_t)row * D_FEAT + k] = feat[(size_t)row * D_FEAT + k] * invn;

    float c = 0.f;
    if (lane < D_CONT) {
        float v = cont[(size_t)row * D_CONT + lane];
        c = v * v;
    }
    #pragma unroll
    for (int o = 16; o >= 1; o >>= 1) c += __shfl_xor(c, o, 32);

    if (lane == 0) {
        csum[row] = c;
        pos[row]  = 0.f;
        den[row]  = 0.f;
    }
}

// ---------------------------------------------------------------------------
// Kernel 2: similarity, 32x32 macro-tile per wave (2x2 register blocking).
//  cat Gram: 84 k-steps x 4 WMMA;  cont Gram: 4 k-steps x 4 WMMA (K=14 padded)
// ---------------------------------------------------------------------------
__global__ void sim_kernel(const float* __restrict__ cont,
                           const float* __restrict__ cat,
                           const float* __restrict__ csum,
                           float* __restrict__ sim) {
    const int lane = threadIdx.x & 31;
    const int wave = threadIdx.x >> 5;
    const int tile = blockIdx.x * 8 + wave;
    if (tile >= N_MT * N_MT) return;
    const int i0 = (tile / N_MT) * 32;
    const int j0 = (tile % N_MT) * 32;

    const int half = lane >> 4;       // selects K pair {k,k+1} vs {k+2,k+3}
    const int l    = lane & 15;
    const int rA0 = i0 + l, rA1 = rA0 + 16;
    const int rB0 = j0 + l, rB1 = rB0 + 16;

    // ---- categorical Gram ----
    v8f c00 = {}, c01 = {}, c10 = {}, c11 = {};
    const float* pa0 = cat + (size_t)rA0 * K_CAT + 2 * half;
    const float* pa1 = cat + (size_t)rA1 * K_CAT + 2 * half;
    const float* pb0 = cat + (size_t)rB0 * K_CAT + 2 * half;
    const float* pb1 = cat + (size_t)rB1 * K_CAT + 2 * half;
    #pragma unroll 4
    for (int k = 0; k < K_CAT; k += 4) {
        v2f a0 = *(const v2f*)(pa0 + k);
        v2f a1 = *(const v2f*)(pa1 + k);
        v2f b0 = *(const v2f*)(pb0 + k);
        v2f b1 = *(const v2f*)(pb1 + k);
        c00 = wmma_f32_4(a0, b0, c00);
        c01 = wmma_f32_4(a0, b1, c01);
        c10 = wmma_f32_4(a1, b0, c10);
        c11 = wmma_f32_4(a1, b1, c11);
    }

    // ---- continuous Gram (K=14 zero-padded to 16) ----
    v8f d00 = {}, d01 = {}, d10 = {}, d11 = {};
    #pragma unroll
    for (int k = 0; k < 16; k += 4) {
        const int ka = k + 2 * half;
        const bool v0 = (ka < D_CONT), v1 = (ka + 1 < D_CONT);
        v2f a0, a1, b0, b1;
        a0.x = v0 ? cont[(size_t)rA0 * D_CONT + ka]     : 0.f;
        a0.y = v1 ? cont[(size_t)rA0 * D_CONT + ka + 1] : 0.f;
        a1.x = v0 ? cont[(size_t)rA1 * D_CONT + ka]     : 0.f;
        a1.y = v1 ? cont[(size_t)rA1 * D_CONT + ka + 1] : 0.f;
        b0.x = v0 ? cont[(size_t)rB0 * D_CONT + ka]     : 0.f;
        b0.y = v1 ? cont[(size_t)rB0 * D_CONT + ka + 1] : 0.f;
        b1.x = v0 ? cont[(size_t)rB1 * D_CONT + ka]     : 0.f;
        b1.y = v1 ? cont[(size_t)rB1 * D_CONT + ka + 1] : 0.f;
        d00 = wmma_f32_4(a0, b0, d00);
        d01 = wmma_f32_4(a0, b1, d01);
        d10 = wmma_f32_4(a1, b0, d10);
        d11 = wmma_f32_4(a1, b1, d11);
    }

    // ---- combine + store (VGPR r -> row base + r + 8*half, col base + l) ----
    const float csC0 = csum[j0 + l];
    const float csC1 = csum[j0 + 16 + l];
    #pragma unroll
    for (int r = 0; r < 8; ++r) {
        const int row0 = i0 + r + 8 * half;
        const int row1 = row0 + 16;
        const int col0 = j0 + l;
        const int col1 = col0 + 16;
        const float csR0 = csum[row0];
        const float csR1 = csum[row1];

        float dd, dist, cs, sval;

        dd   = csR0 + csC0 - 2.f * d00[r];
        dist = sqrtf(fmaxf(dd, 0.f));
        cs   = 2.f / (1.f + dist) - 1.f;
        sval = LAM * cs + (1.f - LAM) * (c00[r] * (1.f / (float)K_CAT));
        if (row0 == col0) sval = 0.f;
        sim[(size_t)row0 * N_ROWS + col0] = sval;

        dd   = csR0 + csC1 - 2.f * d01[r];
        dist = sqrtf(fmaxf(dd, 0.f));
        cs   = 2.f / (1.f + dist) - 1.f;
        sval = LAM * cs + (1.f - LAM) * (c01[r] * (1.f / (float)K_CAT));
        if (row0 == col1) sval = 0.f;
        sim[(size_t)row0 * N_ROWS + col1] = sval;

        dd   = csR1 + csC0 - 2.f * d10[r];
        dist = sqrtf(fmaxf(dd, 0.f));
        cs   = 2.f / (1.f + dist) - 1.f;
        sval = LAM * cs + (1.f - LAM) * (c10[r] * (1.f / (float)K_CAT));
        if (row1 == col0) sval = 0.f;
        sim[(size_t)row1 * N_ROWS + col0] = sval;

        dd   = csR1 + csC1 - 2.f * d11[r];
        dist = sqrtf(fmaxf(dd, 0.f));
        cs   = 2.f / (1.f + dist) - 1.f;
        sval = LAM * cs + (1.f - LAM) * (c11[r] * (1.f / (float)K_CAT));
        if (row1 == col1) sval = 0.f;
        sim[(size_t)row1 * N_ROWS + col1] = sval;
    }
}

// ---------------------------------------------------------------------------
// Kernel 3: per-row max of similarity (one wave per row; sim is L2-resident)
// ---------------------------------------------------------------------------
__global__ void rowmax_kernel(const float* __restrict__ sim,
                              float* __restrict__ rowmax) {
    const int lane = threadIdx.x & 31;
    const int wave = threadIdx.x >> 5;
    const int row  = blockIdx.x * 8 + wave;
    if (row >= N_ROWS) return;

    float m = -1.0e30f;
    for (int c = lane; c < N_ROWS; c += 32)
        m = fmaxf(m, sim[(size_t)row * N_ROWS + c]);
    #pragma unroll
    for (int o = 16; o >= 1; o >>= 1) m = fmaxf(m, __shfl_xor(m, o, 32));
    if (lane == 0) rowmax[row] = m;
}

// ---------------------------------------------------------------------------
// Kernel 4: logits Gram (K=512, 128 k-steps x 4 WMMA, 32x32 macro-tile)
// fused with exp + mask + row reduction + atomic accumulation.
// ---------------------------------------------------------------------------
__global__ void loss_tiles_kernel(const float* __restrict__ fn,
                                  const float* __restrict__ sim,
                                  const float* __restrict__ rowmax,
                                  float* __restrict__ pos,
                                  float* __restrict__ den) {
    const int lane = threadIdx.x & 31;
    const int wave = threadIdx.x >> 5;
    const int tile = blockIdx.x * 8 + wave;
    if (tile >= N_MT * N_MT) return;
    const int i0 = (tile / N_MT) * 32;
    const int j0 = (tile % N_MT) * 32;

    const int half = lane >> 4;
    const int l    = lane & 15;

    v8f c00 = {}, c01 = {}, c10 = {}, c11 = {};
    const float* pa0 = fn + (size_t)(i0 + l) * D_FEAT + 2 * half;
    const float* pa1 = pa0 + (size_t)16 * D_FEAT;
    const float* pb0 = fn + (size_t)(j0 + l) * D_FEAT + 2 * half;
    const float* pb1 = pb0 + (size_t)16 * D_FEAT;
    #pragma unroll 4
    for (int k = 0; k < D_FEAT; k += 4) {
        v2f a0 = *(const v2f*)(pa0 + k);
        v2f a1 = *(const v2f*)(pa1 + k);
        v2f b0 = *(const v2f*)(pb0 + k);
        v2f b1 = *(const v2f*)(pb1 + k);
        c00 = wmma_f32_4(a0, b0, c00);
        c01 = wmma_f32_4(a0, b1, c01);
        c10 = wmma_f32_4(a1, b0, c10);
        c11 = wmma_f32_4(a1, b1, c11);
    }

    #pragma unroll
    for (int r = 0; r < 8; ++r) {
        const int row0 = i0 + r + 8 * half;
        const int row1 = row0 + 16;
        const int col0 = j0 + l;
        const int col1 = col0 + 16;
        const float rm0 = rowmax[row0] - SIM_THR;
        const float rm1 = rowmax[row1] - SIM_THR;

        float e00 = expf(c00[r] * INV_TEMP);
        float e01 = expf(c01[r] * INV_TEMP);
        float e10 = expf(c10[r] * INV_TEMP);
        float e11 = expf(c11[r] * INV_TEMP);
        float s00 = sim[(size_t)row0 * N_ROWS + col0];
        float s01 = sim[(size_t)row0 * N_ROWS + col1];
        float s10 = sim[(size_t)row1 * N_ROWS + col0];
        float s11 = sim[(size_t)row1 * N_ROWS + col1];

        float den0 = e00 + e01;
        float pos0 = (s00 >= rm0 ? e00 : 0.f) + (s01 >= rm0 ? e01 : 0.f);
        float den1 = e10 + e11;
        float pos1 = (s10 >= rm1 ? e10 : 0.f) + (s11 >= rm1 ? e11 : 0.f);

        // reduce across the 16 lanes sharing each row
        #pragma unroll
        for (int o = 8; o >= 1; o >>= 1) {
            den0 += __shfl_xor(den0, o, 16);
            pos0 += __shfl_xor(pos0, o, 16);
            den1 += __shfl_xor(den1, o, 16);
            pos1 += __shfl_xor(pos1, o, 16);
        }
        if (l == 0) {
            atomicAdd(&den[row0], den0);
            atomicAdd(&pos[row0], pos0);
            atomicAdd(&den[row1], den1);
            atomicAdd(&pos[row1], pos1);
        }
    }
}

// ---------------------------------------------------------------------------
// Kernel 5: loss = mean( log(den) - log(pos) )
// ---------------------------------------------------------------------------
__global__ void finalize_kernel(const float* __restrict__ pos,
                                const float* __restrict__ den,
                                float* __restrict__ out) {
    __shared__ float sm[256];
    float acc = 0.f;
    for (int r = threadIdx.x; r < N_ROWS; r += 256)
        acc += logf(den[r]) - logf(pos[r]);
    sm[threadIdx.x] = acc;
    __syncthreads();
    for (int o = 128; o > 0; o >>= 1) {
        if ((int)threadIdx.x < o) sm[threadIdx.x] += sm[threadIdx.x + o];
        __syncthreads();
    }
    if (threadIdx.x == 0) out[0] = sm[0] * (1.0f / (float)N_ROWS);
}

extern "C" void kernel_launch(void* const* d_in, const int* in_sizes, int n_in,
                              void* d_out, int out_size, void* d_ws, size_t ws_size,
                              hipStream_t stream) {
    (void)in_sizes; (void)n_in; (void)out_size; (void)ws_size;
    const float* feat = (const float*)d_in[0];   // [4096, 512]
    const float* cont = (const float*)d_in[1];   // [4096, 14]
    const float* cat  = (const float*)d_in[2];   // [4096, 336]
    float* out = (float*)d_out;                  // scalar

    // workspace layout (~72.1 MB; sim matrix is L2-resident at 64 MB)
    float* sim    = (float*)d_ws;                          // 4096*4096
    float* fn     = sim + (size_t)N_ROWS * N_ROWS;         // 4096*512
    float* csum   = fn + (size_t)N_ROWS * D_FEAT;          // 4096
    float* rowmax = csum + N_ROWS;                         // 4096
    float* pos    = rowmax + N_ROWS;                       // 4096
    float* den    = pos + N_ROWS;                          // 4096

    const int macroTiles = N_MT * N_MT;                    // 16384
    const int tileBlocks = macroTiles / 8;                 // 8 waves/block

    prep_kernel<<<N_ROWS / 8, 256, 0, stream>>>(feat, cont, fn, csum, pos, den);
    sim_kernel<<<tileBlocks, 256, 0, stream>>>(cont, cat, csum, sim);
    rowmax_kernel<<<N_ROWS / 8, 256, 0, stream>>>(sim, rowmax);
    loss_tiles_kernel<<<tileBlocks, 256, 0, stream>>>(fn, sim, rowmax, pos, den);
    finalize_kernel<<<1, 256, 0, stream>>>(pos, den, out);
}